// RandomFeatures_44083544326880
// MI455X (gfx1250) — compile-verified
//
#include <hip/hip_runtime.h>
#include <hip/hip_bf16.h>
#include <stdint.h>

#define B_DIM   32
#define T_DIM   512
#define K_DIM   4096
#define KLEN    11
#define KPB     8       // channels per block
#define THREADS 256

#ifndef __has_builtin
#define __has_builtin(x) 0
#endif

#if defined(__gfx1250__) && \
    __has_builtin(__builtin_amdgcn_global_load_async_to_lds_b128) && \
    __has_builtin(__builtin_amdgcn_s_wait_asynccnt)
#define HAVE_ASYNC_LDS 1
#else
#define HAVE_ASYNC_LDS 0
#endif

typedef int v4i __attribute__((ext_vector_type(4)));
typedef __attribute__((address_space(1))) v4i gv4i;   // global 128-bit chunk
typedef __attribute__((address_space(3))) v4i lv4i;   // LDS 128-bit chunk

__global__ __launch_bounds__(THREADS)
void rocket_conv_feats(const float* __restrict__ x,
                       const float* __restrict__ w,
                       const float* __restrict__ bias,
                       const int*   __restrict__ dil,
                       const int*   __restrict__ pad,
                       const int*   __restrict__ olen,
                       float*       __restrict__ out)
{
    // x staged in LDS: row b stored rotated right by 8*b words so the four
    // batch rows touched by one wave32 occupy disjoint 8-bank ranges.
    __shared__ float xs[B_DIM * T_DIM];   // 64 KB

    const int tid = threadIdx.x;
    const int b   = tid >> 3;          // 0..31
    const int ps  = tid & 7;           // p-phase / staging segment
    const int rot = b * 8;             // word rotation for bank stagger

    // Prefetch this block's weight rows into L2/L0 while we stage x.
    __builtin_prefetch(w + (size_t)blockIdx.x * KPB * KLEN, 0, 1);

    // ---- stage x[b][seg*64 .. seg*64+63] into LDS (rotated) ----
    const float* gsrc = x + b * T_DIM + ps * 64;
#if HAVE_ASYNC_LDS
    #pragma unroll
    for (int i = 0; i < 16; ++i) {
        const int w0 = (ps * 64 + i * 4 + rot) & (T_DIM - 1);   // chunk never wraps (mult of 4)
        gv4i* gp = (gv4i*)(uintptr_t)(gsrc + i * 4);
        lv4i* lp = (lv4i*)(uint32_t)(uintptr_t)(xs + b * T_DIM + w0);
        __builtin_amdgcn_global_load_async_to_lds_b128(gp, lp, 0, 0);
    }
    __builtin_amdgcn_s_wait_asynccnt(0);
    __syncthreads();
#else
    {
        float* ldst = xs + b * T_DIM;
        #pragma unroll
        for (int i = 0; i < 16; ++i) {
            const int w0 = (ps * 64 + i * 4 + rot) & (T_DIM - 1);
            *(float4*)(ldst + w0) = *(const float4*)(gsrc + i * 4);
        }
    }
    __syncthreads();
#endif

    const int rowbase = b * T_DIM;

    for (int kk = 0; kk < KPB; ++kk) {
        const int k  = blockIdx.x * KPB + kk;       // block-uniform -> scalar loads
        const int d  = dil[k];
        const int pd = pad[k];
        const int ol = olen[k];
        const float bs = bias[k];

        float wj[KLEN];
        #pragma unroll
        for (int j = 0; j < KLEN; ++j) wj[j] = w[k * KLEN + j];

        float mx  = -__builtin_inff();
        float cnt = 0.0f;

        // All 11 taps in-bounds iff  pd <= p < T + pd - 10*d  (block-uniform).
        // Expressed in wave-uniform blocks of 8 (p = m*8 + ps, ps = 0..7):
        int lo  = (pd > 0) ? pd : 0;
        int hiv = T_DIM + pd - (KLEN - 1) * d;
        if (hiv > T_DIM) hiv = T_DIM;
        if (hiv < 0)     hiv = 0;
        int mlo = (lo + 7) >> 3;                       // first fully-interior block
        int mhi = (hiv >= 7) ? ((hiv - 7) >> 3) : 0;   // one-past-last interior block
        if (mlo > (T_DIM / 8)) mlo = T_DIM / 8;
        if (mhi > (T_DIM / 8)) mhi = T_DIM / 8;
        if (mhi < mlo)         mhi = mlo;

#define MASKED_BODY(M)                                                          \
        {                                                                       \
            const int p    = (M) * 8 + ps;                                      \
            const int base = p - pd;                                            \
            float acc = bs;                                                     \
            _Pragma("unroll")                                                   \
            for (int j = 0; j < KLEN; ++j) {                                    \
                const int idx = base + j * d;                                   \
                if ((unsigned)idx < (unsigned)T_DIM) {                          \
                    acc = __builtin_fmaf(wj[j],                                 \
                            xs[rowbase + ((idx + rot) & (T_DIM - 1))], acc);    \
                }                                                               \
            }                                                                   \
            if (p < ol) {                                                       \
                mx = fmaxf(mx, acc);                                            \
                cnt += (acc > 0.0f) ? 1.0f : 0.0f;                              \
            }                                                                   \
        }

        for (int m = 0; m < mlo; ++m)       // masked head
            MASKED_BODY(m)

        for (int m = mlo; m < mhi; ++m) {   // unmasked interior (no bound checks)
            const int p = m * 8 + ps;
            float acc = bs;
            int idx = p - pd + rot;
            #pragma unroll
            for (int j = 0; j < KLEN; ++j) {
                acc = __builtin_fmaf(wj[j], xs[rowbase + (idx & (T_DIM - 1))], acc);
                idx += d;
            }
            if (p < ol) {
                mx = fmaxf(mx, acc);
                cnt += (acc > 0.0f) ? 1.0f : 0.0f;
            }
        }

        for (int m = mhi; m < T_DIM / 8; ++m)   // masked tail
            MASKED_BODY(m)
#undef MASKED_BODY

        // 8-lane tree reduction; groups of 8 lanes are aligned inside wave32.
        #pragma unroll
        for (int m = 1; m < 8; m <<= 1) {
            mx   = fmaxf(mx, __shfl_xor(mx, m, 32));
            cnt += __shfl_xor(cnt, m, 32);
        }

        if (ps == 0) {
            out[b * (2 * K_DIM) + 2 * k]     = mx;
            out[b * (2 * K_DIM) + 2 * k + 1] = cnt / (float)ol;
        }
    }
}

extern "C" void kernel_launch(void* const* d_in, const int* in_sizes, int n_in,
                              void* d_out, int out_size, void* d_ws, size_t ws_size,
                              hipStream_t stream) {
    const float* x    = (const float*)d_in[0];
    const float* w    = (const float*)d_in[1];
    const float* bs   = (const float*)d_in[2];
    const int*   dil  = (const int*)d_in[3];
    const int*   pad  = (const int*)d_in[4];
    const int*   olen = (const int*)d_in[5];
    float* out = (float*)d_out;

    dim3 grid(K_DIM / KPB);
    dim3 block(THREADS);
    hipLaunchKernelGGL(rocket_conv_feats, grid, block, 0, stream,
                       x, w, bs, dil, pad, olen, out);
}